// SpectralFractionalConv2d_84267258347799
// MI455X (gfx1250) — compile-verified
//
#include <hip/hip_runtime.h>

typedef float v8f __attribute__((ext_vector_type(8)));
typedef float v2f __attribute__((ext_vector_type(2)));

// ---------------------------------------------------------------------------
// FP32 WMMA: D = A(16x4) * B(4x16) + C(16x16), full precision matrix path.
// ---------------------------------------------------------------------------
static __device__ __forceinline__ v8f wmma4(v2f a, v2f b, v8f c) {
  return __builtin_amdgcn_wmma_f32_16x16x4_f32(
      /*neg_a=*/false, a, /*neg_b=*/false, b,
      /*c_mod=*/(short)0, c, /*reuse_a=*/false, /*reuse_b=*/false);
}

// ---------------------------------------------------------------------------
// DFrFT matrix construction helpers (Candan): S = circ([0,1,0,..,0,1]) +
// diag(2cos(2*pi*k/N)), P = even/odd symmetrizer, CS = P S P^T block-diagonal.
// ---------------------------------------------------------------------------
static __device__ __forceinline__ float S_mat(int a, int b, int N) {
  int d = a - b; if (d < 0) d += N;
  float v = (d == 1 || d == N - 1) ? 1.0f : 0.0f;
  if (a == b) v += 2.0f * cosf(6.28318530717958647692f * (float)a / (float)N);
  return v;
}

static __device__ __forceinline__ int P_row(int i, int N, int cols[2], float vals[2]) {
  const float c = 0.70710678118654752440f;
  int r = N >> 1;
  if (i == 0) { cols[0] = 0; vals[0] = 1.0f; return 1; }
  if (i == r) { cols[0] = r; vals[0] = 1.0f; return 1; }
  if (i < r)  { cols[0] = i; vals[0] = c;  cols[1] = N - i; vals[1] = c; return 2; }
  cols[0] = i; vals[0] = -c; cols[1] = N - i; vals[1] = c; return 2;
}

static __device__ __forceinline__ float CS_elem(int i, int j, int N) {
  int ca[2], cb[2]; float va[2], vb[2];
  int na = P_row(i, N, ca, va);
  int nb = P_row(j, N, cb, vb);
  float acc = 0.0f;
  for (int x = 0; x < na; ++x)
    for (int y = 0; y < nb; ++y)
      acc += va[x] * vb[y] * S_mat(ca[x], cb[y], N);
  return acc;
}

// ---------------------------------------------------------------------------
// Kernel 1: parallel cyclic (chess-tournament) Jacobi eigensolver in LDS.
// Block 0: C2 = CS[0:65,0:65]; Block 1: S2 = CS[65:128,65:128].
// Writes eigenvector columns sorted by DESCENDING eigenvalue.
// ---------------------------------------------------------------------------
__global__ __launch_bounds__(256) void dfrft_jacobi(float* __restrict__ VC,
                                                    float* __restrict__ VS) {
  const int blk = blockIdx.x;
  const int N   = 128;
  const int n   = (blk == 0) ? 65 : 63;
  const int off = (blk == 0) ? 0 : 65;
  const int np  = (n + 1) & ~1;      // 66 or 64 (pad to even)
  const int npairs = np >> 1;        // 33 or 32

  __shared__ float Am[66 * 66];
  __shared__ float Vm[66 * 66];
  __shared__ float rc[33], rs[33];
  __shared__ int   ra[33], rb[33];
  __shared__ int   perm[66];

  const int tid = threadIdx.x;

  for (int t = tid; t < np * np; t += 256) {
    int i = t / np, j = t % np;
    Am[i * 66 + j] = (i < n && j < n) ? CS_elem(off + i, off + j, N) : 0.0f;
    Vm[i * 66 + j] = (i == j) ? 1.0f : 0.0f;
  }
  __syncthreads();

  for (int sweep = 0; sweep < 30; ++sweep) {
    for (int r = 0; r < np - 1; ++r) {
      // Phase A: rotations for all disjoint pairs of this round
      if (tid < npairs) {
        int a, b;
        if (tid == 0) { a = np - 1; b = r % (np - 1); }
        else {
          a = (r + tid) % (np - 1);
          b = (r - tid + 2 * (np - 1)) % (np - 1);
        }
        if (a > b) { int t2 = a; a = b; b = t2; }
        float c = 1.0f, s = 0.0f;
        if (b < n) {
          float apq = Am[a * 66 + b];
          if (fabsf(apq) > 1e-12f) {
            float phi = 0.5f * atan2f(2.0f * apq, Am[b * 66 + b] - Am[a * 66 + a]);
            c = cosf(phi); s = sinf(phi);
          }
        }
        ra[tid] = a; rb[tid] = b; rc[tid] = c; rs[tid] = s;
      }
      __syncthreads();
      // Phase B: row rotations (pairs own disjoint rows)
      for (int t = tid; t < npairs * np; t += 256) {
        int k = t / np, j = t % np;
        int a = ra[k], b = rb[k]; float c = rc[k], s = rs[k];
        float x = Am[a * 66 + j], y = Am[b * 66 + j];
        Am[a * 66 + j] = c * x - s * y;
        Am[b * 66 + j] = s * x + c * y;
      }
      __syncthreads();
      // Phase C: column rotations on A and V (disjoint columns)
      for (int t = tid; t < npairs * np; t += 256) {
        int k = t / np, i2 = t % np;
        int a = ra[k], b = rb[k]; float c = rc[k], s = rs[k];
        float x = Am[i2 * 66 + a], y = Am[i2 * 66 + b];
        Am[i2 * 66 + a] = c * x - s * y;
        Am[i2 * 66 + b] = s * x + c * y;
        x = Vm[i2 * 66 + a]; y = Vm[i2 * 66 + b];
        Vm[i2 * 66 + a] = c * x - s * y;
        Vm[i2 * 66 + b] = s * x + c * y;
      }
      __syncthreads();
    }
  }

  if (tid == 0) {   // selection sort by descending eigenvalue (n <= 65)
    for (int i = 0; i < n; ++i) perm[i] = i;
    for (int i = 0; i < n; ++i) {
      int best = i;
      for (int j = i + 1; j < n; ++j)
        if (Am[perm[j] * 66 + perm[j]] > Am[perm[best] * 66 + perm[best]]) best = j;
      int t2 = perm[i]; perm[i] = perm[best]; perm[best] = t2;
    }
  }
  __syncthreads();

  float* outp = (blk == 0) ? VC : VS;
  for (int t = tid; t < n * n; t += 256) {
    int row = t / n, col = t % n;
    outp[row * n + col] = Vm[row * 66 + perm[col]];
  }
}

// ---------------------------------------------------------------------------
// Kernel 2: assemble E (128x128): even cols <- P@[vc;0], odd cols <- P@[0;vs],
// with numpy's interleave + delete-col-(N-1) layout for even N.
// ---------------------------------------------------------------------------
__global__ void build_E(const float* __restrict__ VC, const float* __restrict__ VS,
                        float* __restrict__ E) {
  const int N = 128;
  int u = blockIdx.x;
  int col = threadIdx.x;
  int src_c, j;
  if (col == 127)            { src_c = 1; j = 64; }           // was col 128 (even)
  else if ((col & 1) == 0)   { src_c = 1; j = col >> 1; }
  else                       { src_c = 0; j = (col - 1) >> 1; }
  int ca[2]; float va[2];
  int na = P_row(u, N, ca, va);
  float acc = 0.0f;
  for (int x = 0; x < na; ++x) {
    int a = ca[x];
    if (src_c) { if (a < 65)  acc += va[x] * VC[a * 65 + j]; }
    else       { if (a >= 65) acc += va[x] * VS[(a - 65) * 63 + j]; }
  }
  E[u * 128 + col] = acc;
}

// ---------------------------------------------------------------------------
// Kernel 3: A[u][v] = sum_j E[u][j] * E[v][j] * exp(-i*(pi/4)*k_j),
// k = [0..126, 128], alpha = 0.5. A is symmetric by construction.
// ---------------------------------------------------------------------------
__global__ void build_A(const float* __restrict__ E,
                        float* __restrict__ Are, float* __restrict__ Aim) {
  int u = blockIdx.x, v = threadIdx.x;
  float sr = 0.0f, si = 0.0f;
  for (int j = 0; j < 128; ++j) {
    int kj = (j == 127) ? 128 : j;
    float ang = -0.78539816339744830962f * (float)kj;   // -(pi/4)*k
    float ss, cc; __sincosf(ang, &ss, &cc);
    float e = E[u * 128 + j] * E[v * 128 + j];
    sr += e * cc; si += e * ss;
  }
  Are[u * 128 + v] = sr;
  Aim[u * 128 + v] = si;
}

// ---------------------------------------------------------------------------
// Batched complex GEMM via V_WMMA_F32_16X16X4_F32, specialized at compile
// time on <HASPIM, HASBIAS> so the hot loop is branch-free straight-line WMMA.
// C = P * Q (+bias per row), planes split re/im; HASPIM=false => P real.
// pis/qis: sign applied to imag plane at load (conjugation).
// Requires M%64==0, N%128==0, K%16==0 (true for all stages here).
// Block: 256 threads = 8 wave32, block tile 64(M) x 128(N), BK=16.
// ---------------------------------------------------------------------------
#define BM 64
#define BN 128
#define BK 16

template <bool HASPIM, bool HASBIAS>
__global__ __launch_bounds__(256) void cgemm_wmma(
    const float* __restrict__ Pre, const float* __restrict__ Pim,
    const float* __restrict__ Qre, const float* __restrict__ Qim,
    float* __restrict__ Cre, float* __restrict__ Cim,
    int M, int N, int K,
    long long sP, long long sQ, long long sC,
    float pis, float qis,
    const float* __restrict__ bre, const float* __restrict__ bim) {
  const long long batch = blockIdx.z;
  Pre += batch * sP; if (HASPIM) Pim += batch * sP;
  Qre += batch * sQ; Qim += batch * sQ;
  Cre += batch * sC; Cim += batch * sC;

  __shared__ float sAre[BM][BK + 1];
  __shared__ float sAim[BM][BK + 1];
  __shared__ float sBre[BK][BN + 4];
  __shared__ float sBim[BK][BN + 4];

  const int tid  = threadIdx.x;
  const int lane = tid & 31;
  const int wave = tid >> 5;
  const int wm   = (wave >> 2) * 32;     // 0 or 32
  const int wn   = (wave & 3) * 32;      // 0,32,64,96
  const int bm0  = blockIdx.y * BM;
  const int bn0  = blockIdx.x * BN;

  v8f accRe[2][2], accIm[2][2];
#pragma unroll
  for (int i = 0; i < 2; ++i)
#pragma unroll
    for (int j = 0; j < 2; ++j) { accRe[i][j] = (v8f){}; accIm[i][j] = (v8f){}; }

  for (int k0 = 0; k0 < K; k0 += BK) {
    for (int t = tid; t < BM * BK; t += 256) {       // 4 elems/thread
      int r = t >> 4, c = t & 15;
      long long g = (long long)(bm0 + r) * K + (k0 + c);
      sAre[r][c] = Pre[g];
      if (HASPIM) sAim[r][c] = pis * Pim[g];
    }
    for (int t = tid; t < BK * BN; t += 256) {       // 8 elems/thread
      int r = t >> 7, c = t & 127;
      long long g = (long long)(k0 + r) * N + (bn0 + c);
      sBre[r][c] = Qre[g];
      sBim[r][c] = qis * Qim[g];
    }
    __syncthreads();

    const int mrow0 = wm + (lane & 15);
    const int ncol0 = wn + (lane & 15);
    const int khalf = (lane >> 4) << 1;              // 0 or 2

#pragma unroll
    for (int kk = 0; kk < BK; kk += 4) {
      const int kb = kk + khalf;
      v2f ar[2], ai[2], br[2], bi[2];
#pragma unroll
      for (int tm = 0; tm < 2; ++tm) {
        int row = mrow0 + tm * 16;
        ar[tm] = (v2f){sAre[row][kb], sAre[row][kb + 1]};
        if (HASPIM) ai[tm] = (v2f){sAim[row][kb], sAim[row][kb + 1]};
      }
#pragma unroll
      for (int tn = 0; tn < 2; ++tn) {
        int col = ncol0 + tn * 16;
        br[tn] = (v2f){sBre[kb][col], sBre[kb + 1][col]};
        bi[tn] = (v2f){sBim[kb][col], sBim[kb + 1][col]};
      }
#pragma unroll
      for (int tm = 0; tm < 2; ++tm)
#pragma unroll
        for (int tn = 0; tn < 2; ++tn) {
          accRe[tm][tn] = wmma4(ar[tm], br[tn], accRe[tm][tn]);
          accIm[tm][tn] = wmma4(ar[tm], bi[tn], accIm[tm][tn]);
          if (HASPIM) {
            v2f nai = -ai[tm];                       // f32 WMMA has no A-neg bit
            accRe[tm][tn] = wmma4(nai, bi[tn], accRe[tm][tn]);
            accIm[tm][tn] = wmma4(ai[tm], br[tn], accIm[tm][tn]);
          }
        }
    }
    __syncthreads();
  }

#pragma unroll
  for (int tm = 0; tm < 2; ++tm)
#pragma unroll
    for (int tn = 0; tn < 2; ++tn)
#pragma unroll
      for (int j = 0; j < 8; ++j) {
        int gm = bm0 + wm + tm * 16 + j + ((lane >> 4) << 3);
        int gn = bn0 + wn + tn * 16 + (lane & 15);
        float re = accRe[tm][tn][j];
        float im = accIm[tm][tn][j];
        if (HASBIAS) { re += bre[gm]; im += bim[gm]; }
        long long g = (long long)gm * N + gn;
        Cre[g] = re;
        Cim[g] = im;
      }
}

// ---------------------------------------------------------------------------
// Launch: build A once, then 5 batched complex GEMM stages (memory-bound:
// ~0.75 GB through HBM -> ~32us at 23.3 TB/s; 70 GFLOP is trivial for WMMA).
// ---------------------------------------------------------------------------
extern "C" void kernel_launch(void* const* d_in, const int* in_sizes, int n_in,
                              void* d_out, int out_size, void* d_ws, size_t ws_size,
                              hipStream_t stream) {
  (void)in_sizes; (void)n_in; (void)out_size; (void)ws_size;
  const float* x  = (const float*)d_in[0];   // (16,64,128,128)
  const float* wr = (const float*)d_in[1];   // (64,64)
  const float* wi = (const float*)d_in[2];   // (64,64)
  const float* br = (const float*)d_in[3];   // (64,)
  const float* bi = (const float*)d_in[4];   // (64,)
  float* out = (float*)d_out;                // (2,16,64,128,128)

  float* ws   = (float*)d_ws;
  float* Are  = ws;                 // 16384
  float* Aim  = ws + 16384;         // 16384
  float* Ebuf = ws + 32768;         // 16384
  float* VC   = ws + 49152;         // 65*65
  float* VS   = ws + 53377;         // 63*63
  const long long IMG   = 16384;            // 128*128
  const long long PLANE = 1024 * IMG;       // B*C*H*W = 16,777,216
  float* T1re = ws + 65536;
  float* T1im = T1re + PLANE;
  float* T2re = T1im + PLANE;
  float* T2im = T2re + PLANE;

  // ---- build DFrFT matrix A (symmetric 128x128 complex) ----
  dfrft_jacobi<<<2, 256, 0, stream>>>(VC, VS);
  build_E<<<128, 128, 0, stream>>>(VC, VS, Ebuf);
  build_A<<<128, 128, 0, stream>>>(Ebuf, Are, Aim);

  const dim3 blk(256);
  // Stage 1a: T1 = X * A           (per image; X real)
  cgemm_wmma<false, false><<<dim3(1, 2, 1024), blk, 0, stream>>>(
      x, nullptr, Are, Aim, T1re, T1im,
      128, 128, 128, IMG, 0LL, IMG, 1.0f, 1.0f, nullptr, nullptr);
  // Stage 1b: T2 = A * T1          (x_ft = A X A^T, A symmetric)
  cgemm_wmma<true, false><<<dim3(1, 2, 1024), blk, 0, stream>>>(
      Are, Aim, T1re, T1im, T2re, T2im,
      128, 128, 128, 0LL, IMG, IMG, 1.0f, 1.0f, nullptr, nullptr);
  // Stage 2: out_ft = W * x_ft + b (per b: 64x64 times 64x16384), into T1
  cgemm_wmma<true, true><<<dim3(128, 1, 16), blk, 0, stream>>>(
      wr, wi, T2re, T2im, T1re, T1im,
      64, 16384, 64, 0LL, 64LL * 16384, 64LL * 16384, 1.0f, 1.0f, br, bi);
  // Stage 3a: U = conj(A) * out_ft (per image), into T2
  cgemm_wmma<true, false><<<dim3(1, 2, 1024), blk, 0, stream>>>(
      Are, Aim, T1re, T1im, T2re, T2im,
      128, 128, 128, 0LL, IMG, IMG, -1.0f, 1.0f, nullptr, nullptr);
  // Stage 3b: Y = U * conj(A) -> out (real plane, imag plane)
  cgemm_wmma<true, false><<<dim3(1, 2, 1024), blk, 0, stream>>>(
      T2re, T2im, Are, Aim, out, out + PLANE,
      128, 128, 128, IMG, 0LL, IMG, 1.0f, -1.0f, nullptr, nullptr);
}